// BarycentricCoordinates_53266184405263
// MI455X (gfx1250) — compile-verified
//
#include <hip/hip_runtime.h>
#include <hip/hip_bf16.h>
#include <math.h>

// BarycentricCoordinates on MI455X (gfx1250, wave32).
// One wave per (v, r, a) problem; N=32 projections == 32 lanes.
//   - stable argsort via per-lane rank (readlane broadcasts, no DS pipe)
//   - inverse permutation via ds_permute_b32 (LDS pipe, no LDS memory)
//   - dot01 Gram (32x32 = v0.v0^T, K padded 2->4) via 4x v_wmma_f32_16x16x4_f32,
//     consumed IN the WMMA D-layout (no transpose-back shuffles):
//       tile(tr,tc) VGPR j, lane l  <->  pair n=16tr+j+8*(l>=16), m=16tc+(l&15)
//   - per-pair scoring with exact-cancellation diagonal, IEEE f32 divides
//   - argmin: local (score,flat) w/ flat tie-break == first occurrence, then
//     5-step shfl_xor butterfly on 2 values; winner lane derived from flat,
//     w-triple fetched with 3 readlanes.

typedef float v2f __attribute__((ext_vector_type(2)));
typedef float v8f __attribute__((ext_vector_type(8)));

#ifndef __has_builtin
#define __has_builtin(x) 0
#endif

#if __has_builtin(__builtin_amdgcn_wmma_f32_16x16x4_f32)
#define USE_WMMA_F32X4 1
#else
#define USE_WMMA_F32X4 0
#endif

#define EPSF 1e-10f

__device__ __forceinline__ float rdlane_f(float x, int uniformLane) {
    return __int_as_float(__builtin_amdgcn_readlane(__float_as_int(x), uniformLane));
}
__device__ __forceinline__ float rdfirst_f(float x) {
    return __int_as_float(__builtin_amdgcn_readfirstlane(__float_as_int(x)));
}

__global__ __launch_bounds__(256)
void BarycentricCoordinates_53266184405263_kernel(const float* __restrict__ tmpl,
                                                  const float* __restrict__ proj,
                                                  float* __restrict__ out,
                                                  int RA, int total) {
    const int lane = threadIdx.x & 31;
    const int q = (blockIdx.x * blockDim.x + threadIdx.x) >> 5;
    if (q >= total) return;                 // whole-wave uniform: EXEC stays all-ones

    const int v  = q / RA;
    const int ra = q - v * RA;

    // Template point (uniform) and this lane's projection point.
    const float tx = tmpl[ra * 2 + 0];
    const float ty = tmpl[ra * 2 + 1];
    const float* pp = proj + (size_t)v * 64;          // 32 points * 2 floats
    const float px = pp[lane * 2 + 0];
    const float py = pp[lane * 2 + 1];

    // Distance (sqrt matches jnp.linalg.norm ordering, incl. rounding ties).
    const float ddx = tx - px, ddy = ty - py;
    const float d = sqrtf(ddx * ddx + ddy * ddy);

    // Stable rank via SGPR broadcasts (VALU readlane, keeps DS pipe free).
    int rank = 0;
#pragma unroll
    for (int k = 0; k < 32; ++k) {
        const float dk = rdlane_f(d, k);
        rank += (dk < d || (dk == d && k < lane)) ? 1 : 0;
    }

    // Inverse permutation via forward scatter: order[rank_j] = j.
    const int ord = __builtin_amdgcn_ds_permute(rank << 2, lane);

    // Gather sorted point; closest = sorted[0].
    const float sx = __shfl(px, ord, 32);
    const float sy = __shfl(py, ord, 32);
    const float cx = rdfirst_f(sx);
    const float cy = rdfirst_f(sy);

    const float v0x = sx - cx, v0y = sy - cy;         // lane i holds v0_i (sorted)
    const float v2x = tx - cx, v2y = ty - cy;

    const float dot00 = v0x * v0x + v0y * v0y;        // per-lane scalars
    const float dot02 = v0x * v2x + v0y * v2y;

    const int  hlane = lane & 15;
    const bool lo    = lane < 16;

#if USE_WMMA_F32X4
    // A 16x4 f32 operand: lanes 0-15 hold row M=lane {K0,K1}; lanes 16-31 {K2,K3}=0.
    // B 4x16 f32 operand has identical contents for the symmetric Gram.
    const float ax0 = __shfl(v0x, hlane, 32);
    const float ay0 = __shfl(v0y, hlane, 32);
    const float ax1 = __shfl(v0x, hlane + 16, 32);
    const float ay1 = __shfl(v0y, hlane + 16, 32);
    v2f A0; A0.x = lo ? ax0 : 0.0f; A0.y = lo ? ay0 : 0.0f;   // points 0..15
    v2f A1; A1.x = lo ? ax1 : 0.0f; A1.y = lo ? ay1 : 0.0f;   // points 16..31
    v8f Z = {};
    v8f G00 = __builtin_amdgcn_wmma_f32_16x16x4_f32(false, A0, false, A0, (short)0, Z, false, false);
    v8f G01 = __builtin_amdgcn_wmma_f32_16x16x4_f32(false, A0, false, A1, (short)0, Z, false, false);
    v8f G10 = __builtin_amdgcn_wmma_f32_16x16x4_f32(false, A1, false, A0, (short)0, Z, false, false);
    v8f G11 = __builtin_amdgcn_wmma_f32_16x16x4_f32(false, A1, false, A1, (short)0, Z, false, false);
#endif

    // m-side broadcasts (one-time): m = 16*tc + (lane&15).
    const float d00mA = __shfl(dot00, hlane, 32);        // tc = 0
    const float d00mB = __shfl(dot00, hlane + 16, 32);   // tc = 1
    const float d02mA = __shfl(dot02, hlane, 32);
    const float d02mB = __shfl(dot02, hlane + 16, 32);
#if !USE_WMMA_F32X4
    const float v0xmA = __shfl(v0x, hlane, 32);
    const float v0xmB = __shfl(v0x, hlane + 16, 32);
    const float v0ymA = __shfl(v0y, hlane, 32);
    const float v0ymB = __shfl(v0y, hlane + 16, 32);
#endif

    // Per-lane pair sweep in the WMMA output layout.
    float best = INFINITY;
    int   bestFlat = 0x7FFFFFFF;            // flat tie-break == first occurrence
    float bw0 = 0.0f, bw1 = 0.0f, bw2 = 0.0f;
    const int hsel = (lane & 16) >> 1;      // +8 rows for lanes 16-31

#pragma unroll
    for (int tr = 0; tr < 2; ++tr) {
#pragma unroll
        for (int j = 0; j < 8; ++j) {
            const int n   = 16 * tr + j + hsel;          // this lane's row
            const float d00n = __shfl(dot00, n, 32);
            const float d02n = __shfl(dot02, n, 32);
#if !USE_WMMA_F32X4
            const float v0xn = __shfl(v0x, n, 32);
            const float v0yn = __shfl(v0y, n, 32);
#endif
#pragma unroll
            for (int tc = 0; tc < 2; ++tc) {
                const int m = 16 * tc + hlane;           // this lane's col
#if USE_WMMA_F32X4
                float dot01 = (tr == 0) ? (tc == 0 ? G00[j] : G01[j])
                                        : (tc == 0 ? G10[j] : G11[j]);
#else
                float dot01 = v0xn * (tc ? v0xmB : v0xmA) + v0yn * (tc ? v0ymB : v0ymA);
#endif
                // Diagonal must cancel exactly like the reference.
                if (n == m) dot01 = d00n;

                const float d00m = tc ? d00mB : d00mA;
                const float d02m = tc ? d02mB : d02mA;

                float den = d00n * d00m - dot01 * dot01;
                den = (den == 0.0f) ? EPSF : den;
                const float p2 = (d00m * d02n - dot01 * d02m) / den;
                const float p1 = (d00n * d02m - dot01 * d02n) / den;
                const float p0 = 1.0f - p2 - p1;

                float sc = fmaxf(fabsf(p0), fmaxf(fabsf(p2), fabsf(p1)))
                         + fmaxf(-p0, 0.0f) + fmaxf(-p2, 0.0f) + fmaxf(-p1, 0.0f);
                // score >= 0 for finite w; argmin prefers NaN -> canonicalize NaN to -1.
                sc = (sc != sc) ? -1.0f : sc;

                const int flat = (n << 5) | m;
                const bool take = (sc < best) || (sc == best && flat < bestFlat);
                best     = take ? sc : best;
                bestFlat = take ? flat : bestFlat;
                bw0 = take ? p0 : bw0;
                bw1 = take ? p2 : bw1;
                bw2 = take ? p1 : bw2;
            }
        }
    }

    // Cross-lane argmin on (score, flat) only.
#pragma unroll
    for (int off = 16; off > 0; off >>= 1) {
        const float os = __shfl_xor(best, off, 32);
        const int   of = __shfl_xor(bestFlat, off, 32);
        const bool t = (os < best) || (os == best && of < bestFlat);
        best     = t ? os : best;
        bestFlat = t ? of : bestFlat;
    }

    // Winner lane is derivable from the flat index in this layout.
    const int nW = bestFlat >> 5;
    const int mW = bestFlat & 31;
    const int lstar = __builtin_amdgcn_readfirstlane(((nW & 8) << 1) | (mW & 15));
    const float fw0 = rdlane_f(bw0, lstar);
    const float fw1 = rdlane_f(bw1, lstar);
    const float fw2 = rdlane_f(bw2, lstar);

    const int rowU = __builtin_amdgcn_readfirstlane(nW);
    const int colU = __builtin_amdgcn_readfirstlane(mW);
    const int ordRow = __builtin_amdgcn_readlane(ord, rowU);
    const int ordCol = __builtin_amdgcn_readlane(ord, colU);
    const int ord0   = __builtin_amdgcn_readfirstlane(ord);

    if (lane == 0) {
        float* iw = out + (size_t)q * 3;
        iw[0] = fw0; iw[1] = fw1; iw[2] = fw2;
        float* ix = out + (size_t)total * 3 + (size_t)q * 3;
        ix[0] = (float)ord0;
        ix[1] = (float)ordRow;
        ix[2] = (float)ordCol;
    }
}

extern "C" void kernel_launch(void* const* d_in, const int* in_sizes, int n_in,
                              void* d_out, int out_size, void* d_ws, size_t ws_size,
                              hipStream_t stream) {
    (void)n_in; (void)out_size; (void)d_ws; (void)ws_size;
    const float* tmpl = (const float*)d_in[0];   // (R, A, 2) float32
    const float* proj = (const float*)d_in[1];   // (V, 32, 2) float32

    const int RA    = in_sizes[0] / 2;           // R*A flattened
    const int V     = in_sizes[1] / 64;          // N=32 points * 2 coords
    const int total = V * RA;                    // one wave32 per problem

    const int threads = 256;                     // 8 waves per block
    const int wavesPerBlock = threads / 32;
    const int blocks = (total + wavesPerBlock - 1) / wavesPerBlock;

    BarycentricCoordinates_53266184405263_kernel<<<blocks, threads, 0, stream>>>(
        tmpl, proj, (float*)d_out, RA, total);
}